// PRGNN_6665789243916
// MI455X (gfx1250) — compile-verified
//
#include <hip/hip_runtime.h>
#include <hip/hip_bf16.h>
#include <math.h>

typedef float v2f __attribute__((ext_vector_type(2)));
typedef float v8f __attribute__((ext_vector_type(8)));

#define N_NODES 50000
#define N_EDGES 800000
#define NB      64
#define F_INV   128
#define CCH     128

// ---------------------------------------------------------------------------
// WMMA f32 GEMM: H[r,c] = sum_k X[r,k] * W[k,c] + bias[c]   (optional relu)
// One wave computes one 16x16 tile via V_WMMA_F32_16X16X4_F32.
// K and NCOLS are compile-time so every load offset is an immediate; the
// wave's B tile (its 16-col strip of W across all K) is hoisted into
// registers (K/4 * 2 VGPRs), leaving one global_load_b64 (A) per WMMA.
// blockDim = (32, NCOLS/16); gridDim.x = Nrows/16. EXEC all-ones per wave.
// A layout: lane l -> row (l&15), VGPR0/1 = K = k0+2*(l>>4) + {0,1}
// B layout: lane l -> col (l&15), VGPR0/1 = same K pair
// C/D layout: VGPR v, lane l -> row v + 8*(l>>4), col (l&15)
// ---------------------------------------------------------------------------
template <int K, int NCOLS>
__global__ __launch_bounds__(256) void wmma_gemm_bias(
    const float* __restrict__ X, const float* __restrict__ W,
    const float* __restrict__ bias, float* __restrict__ H, int relu)
{
  constexpr int KSTEPS = K / 4;
  const int lane = threadIdx.x;        // 0..31
  const int half = lane >> 4;          // 0 or 1
  const int l16  = lane & 15;
  const int row0 = blockIdx.x << 4;
  const int col0 = threadIdx.y << 4;

  const float* xrow = X + (long long)(row0 + l16) * K;
  const float* wcol = W + col0 + l16;

  // Hoist the loop-invariant B tile into registers (immediate offsets).
  v2f breg[KSTEPS];
#pragma unroll
  for (int s = 0; s < KSTEPS; ++s) {
    const int ka = 4 * s + (half << 1);
    breg[s].x = wcol[ka * NCOLS];
    breg[s].y = wcol[(ka + 1) * NCOLS];
  }

  v8f acc = {};
#pragma unroll
  for (int s = 0; s < KSTEPS; ++s) {
    const int ka = 4 * s + (half << 1);
    v2f a;
    a.x = xrow[ka];          // contiguous pair -> global_load_b64, imm offset
    a.y = xrow[ka + 1];
    acc = __builtin_amdgcn_wmma_f32_16x16x4_f32(
        /*neg_a=*/false, a, /*neg_b=*/false, breg[s],
        /*c_mod=*/(short)0, acc, /*reuse_a=*/false, /*reuse_b=*/false);
  }

  const float bb = bias[col0 + l16];
#pragma unroll
  for (int v = 0; v < 8; ++v) {
    const int r = row0 + v + (half << 3);
    float val = acc[v] + bb;
    if (relu) val = fmaxf(val, 0.0f);
    H[(long long)r * NCOLS + col0 + l16] = val;
  }
}

// ---------------------------------------------------------------------------
// Edge scatter: agg[dst[e], :] += h[src[e], :]
// One wave per edge; lane handles CDIM/32 contiguous floats (coalesced
// 512B/256B gather per wave, L2-resident f32 atomics).
// ---------------------------------------------------------------------------
template <int CDIM>
__global__ __launch_bounds__(256) void scatter_kernel(
    const float* __restrict__ h, const int* __restrict__ src,
    const int* __restrict__ dst, float* __restrict__ agg, int E)
{
  constexpr int PER = CDIM / 32;
  const int e = blockIdx.x * blockDim.y + threadIdx.y;
  if (e >= E) return;
  const int s = src[e];
  const int d = dst[e];
  const float* hp = h + (long long)s * CDIM + threadIdx.x * PER;
  float* ap = agg + (long long)d * CDIM + threadIdx.x * PER;
  float v[PER];
#pragma unroll
  for (int i = 0; i < PER; ++i) v[i] = hp[i];
#pragma unroll
  for (int i = 0; i < PER; ++i) atomicAdd(ap + i, v[i]);
}

// ---------------------------------------------------------------------------
// BN (inference, mean=0, var=1) + relu, in place.  y = relu(x*g/sqrt(1+eps)+be)
// ---------------------------------------------------------------------------
__global__ void bn_relu_kernel(float* __restrict__ x, const float* __restrict__ g,
                               const float* __restrict__ be, int total, int cmask)
{
  const int tid = blockIdx.x * blockDim.x + threadIdx.x;
  if (tid >= total) return;
  const float inv = 1.0f / sqrtf(1.0f + 1e-3f);   // folded to a constant
  const int c = tid & cmask;
  const float val = x[tid] * (g[c] * inv) + be[c];
  x[tid] = fmaxf(val, 0.0f);
}

// ---------------------------------------------------------------------------
// Segment mean pooling (C=64): sum + count via atomics, then divide.
// ---------------------------------------------------------------------------
__global__ void pool_sum_kernel(const float* __restrict__ x, const int* __restrict__ seg,
                                float* __restrict__ pool, float* __restrict__ cnt, int N)
{
  const int tid = blockIdx.x * blockDim.x + threadIdx.x;
  if (tid >= N * 64) return;
  const int n = tid >> 6;
  const int c = tid & 63;
  const int s = seg[n];
  atomicAdd(&pool[s * 64 + c], x[tid]);
  if (c == 0) atomicAdd(&cnt[s], 1.0f);
}

__global__ void pool_div_kernel(const float* __restrict__ pool, const float* __restrict__ cnt,
                                float* __restrict__ din, int colOff)
{
  const int tid = blockIdx.x * blockDim.x + threadIdx.x;
  if (tid >= NB * 64) return;
  const int b = tid >> 6;
  const int c = tid & 63;
  din[b * CCH + colOff + c] = pool[tid] / fmaxf(cnt[b], 1.0f);
}

// ---------------------------------------------------------------------------
// Output head: sigmoid(din @ Wo + bo) -> [64,1]
// ---------------------------------------------------------------------------
__global__ void head_kernel(const float* __restrict__ din, const float* __restrict__ Wo,
                            const float* __restrict__ bo, float* __restrict__ out)
{
  const int b = threadIdx.x;
  if (b >= NB) return;
  float s = 0.0f;
#pragma unroll 8
  for (int k = 0; k < CCH; ++k) s += din[b * CCH + k] * Wo[k];
  s += bo[0];
  out[b] = 1.0f / (1.0f + expf(-s));
}

// ---------------------------------------------------------------------------
extern "C" void kernel_launch(void* const* d_in, const int* in_sizes, int n_in,
                              void* d_out, int out_size, void* d_ws, size_t ws_size,
                              hipStream_t stream)
{
  const float* x[2]   = {(const float*)d_in[0],  (const float*)d_in[1]};
  const int*   edg[2] = {(const int*)d_in[2],    (const int*)d_in[3]};
  const int*   seg[2] = {(const int*)d_in[4],    (const int*)d_in[5]};
  const float* Wa[2]  = {(const float*)d_in[6],  (const float*)d_in[14]};
  const float* ba[2]  = {(const float*)d_in[7],  (const float*)d_in[15]};
  const float* ga[2]  = {(const float*)d_in[8],  (const float*)d_in[16]};
  const float* bea[2] = {(const float*)d_in[9],  (const float*)d_in[17]};
  const float* Wb[2]  = {(const float*)d_in[10], (const float*)d_in[18]};
  const float* bb[2]  = {(const float*)d_in[11], (const float*)d_in[19]};
  const float* gb[2]  = {(const float*)d_in[12], (const float*)d_in[20]};
  const float* beb[2] = {(const float*)d_in[13], (const float*)d_in[21]};
  const float* Wd = (const float*)d_in[22];
  const float* bd = (const float*)d_in[23];
  const float* Wo = (const float*)d_in[24];
  const float* bo = (const float*)d_in[25];

  // Workspace layout (floats): total ~12.82M floats (~51.3 MB)
  float* ws    = (float*)d_ws;
  float* bufH  = ws;                                  // N*128 : GEMM out / (h2|agg2)
  float* bufA  = bufH + (size_t)N_NODES * CCH;        // N*128 : agg layer a (in-place BN)
  float* poolS = bufA + (size_t)N_NODES * CCH;        // 64*64
  float* cnt   = poolS + NB * 64;                     // 64 (contiguous after poolS)
  float* din   = cnt + NB;                            // 64*128
  float* dtmp  = din + NB * CCH;                      // 64*128

  for (int br = 0; br < 2; ++br) {
    const int* src = edg[br];
    const int* dst = edg[br] + N_EDGES;

    // ---- GeneralConv a: GEMM 50000x128x128 -> scatter -> BN+relu
    wmma_gemm_bias<128, 128><<<dim3(N_NODES / 16, 1, 1), dim3(32, 8, 1), 0, stream>>>(
        x[br], Wa[br], ba[br], bufH, 0);
    hipMemsetAsync(bufA, 0, (size_t)N_NODES * CCH * sizeof(float), stream);
    scatter_kernel<128><<<N_EDGES / 8, dim3(32, 8, 1), 0, stream>>>(
        bufH, src, dst, bufA, N_EDGES);
    bn_relu_kernel<<<(N_NODES * CCH + 255) / 256, 256, 0, stream>>>(
        bufA, ga[br], bea[br], N_NODES * CCH, CCH - 1);

    // ---- GeneralConv b: GEMM 50000x128x64 -> scatter -> BN+relu
    float* h2   = bufH;                       // reuse: N*64
    float* agg2 = bufH + (size_t)N_NODES * 64;
    wmma_gemm_bias<128, 64><<<dim3(N_NODES / 16, 1, 1), dim3(32, 4, 1), 0, stream>>>(
        bufA, Wb[br], bb[br], h2, 0);
    hipMemsetAsync(agg2, 0, (size_t)N_NODES * 64 * sizeof(float), stream);
    scatter_kernel<64><<<N_EDGES / 8, dim3(32, 8, 1), 0, stream>>>(
        h2, src, dst, agg2, N_EDGES);
    bn_relu_kernel<<<(N_NODES * 64 + 255) / 256, 256, 0, stream>>>(
        agg2, gb[br], beb[br], N_NODES * 64, 63);

    // ---- Segment mean pool -> din[:, br*64 : br*64+64]
    hipMemsetAsync(poolS, 0, (size_t)(NB * 64 + NB) * sizeof(float), stream);
    pool_sum_kernel<<<(N_NODES * 64 + 255) / 256, 256, 0, stream>>>(
        agg2, seg[br], poolS, cnt, N_NODES);
    pool_div_kernel<<<(NB * 64 + 255) / 256, 256, 0, stream>>>(
        poolS, cnt, din, br * 64);
  }

  // ---- Shared Dense(128, relu) x2 on [64,128], via the same WMMA GEMM
  wmma_gemm_bias<128, 128><<<dim3(NB / 16, 1, 1), dim3(32, 8, 1), 0, stream>>>(
      din, Wd, bd, dtmp, 1);
  wmma_gemm_bias<128, 128><<<dim3(NB / 16, 1, 1), dim3(32, 8, 1), 0, stream>>>(
      dtmp, Wd, bd, din, 1);

  // ---- Sigmoid head -> d_out[64]
  head_kernel<<<1, 64, 0, stream>>>(din, Wo, bo, (float*)d_out);
}